// IntegerLookup_72507637891122
// MI455X (gfx1250) — compile-verified
//
#include <hip/hip_runtime.h>
#include <stdint.h>

typedef unsigned long long u64;
#define EMPTY64 0xFFFFFFFFFFFFFFFFull

// ---------------------------------------------------------------------------
// Open-addressing hash table: slot = (value << 32) | key(low 32).
// During insert, value = first-occurrence flat index (u64 atomicMin resolves
// "first").  After the rank pass, value is overwritten with the final rank.
// EMPTY = all-ones; safe even for key==-1 since any real index/rank < 2^31.
// Table is sized to stay resident in MI455X's 192MB L2 (32MB default).
// ---------------------------------------------------------------------------

__device__ __forceinline__ uint32_t hash_key(uint32_t k, uint32_t mask) {
    uint32_t h = k * 2654435761u;
    h ^= h >> 16;
    return h & mask;
}

__device__ __forceinline__ uint32_t lookup_val(const u64* __restrict__ tab,
                                               uint32_t mask, uint32_t ukey) {
    uint32_t s = hash_key(ukey, mask);
    for (uint32_t it = 0; it <= mask; ++it) {
        u64 cur = tab[s];
        if (cur == EMPTY64) return 0xFFFFFFFFu;           // not found
        if ((uint32_t)cur == ukey) return (uint32_t)(cur >> 32);
        s = (s + 1) & mask;
    }
    return 0xFFFFFFFFu;
}

// ---------------------------------------------------------------------------
// Stage 4 consecutive int32 keys per lane into LDS via the CDNA5 async copy
// engine (GLOBAL_LOAD_ASYNC_TO_LDS_B128, tracked by ASYNCcnt).  The low 32
// bits of the flat shared-variable address are the wave-relative LDS byte
// offset (flat->LDS mapping truncates to addr[31:0], ISA 10.2).
// ---------------------------------------------------------------------------
__device__ __forceinline__ void stage4_async(const int* __restrict__ x, int n,
                                             int i0, int* tilePtr) {
    if (i0 + 4 <= n) {
        uint32_t ldsOff = (uint32_t)(uintptr_t)tilePtr;
        u64 gaddr = (u64)(uintptr_t)(x + i0);
        asm volatile("global_load_async_to_lds_b128 %0, %1, off"
                     :: "v"(ldsOff), "v"(gaddr) : "memory");
    } else {
        for (int k = 0; k < 4; ++k)
            tilePtr[k] = (i0 + k < n) ? x[i0 + k] : 0;
    }
    asm volatile("s_wait_asynccnt 0" ::: "memory");       // wave-local completion
}

// ---------------------------------------------------------------------------
__global__ __launch_bounds__(256) void k_init(u64* tab, uint32_t slots) {
    uint32_t i = blockIdx.x * 256u + threadIdx.x;
    uint32_t stride = gridDim.x * 256u;
    for (uint32_t s = i; s < slots; s += stride) tab[s] = EMPTY64;
}

// ---------------------------------------------------------------------------
// Insert pass: claim empty slots with CAS, resolve same-key collisions with
// GLOBAL_ATOMIC_MIN_U64 (index lives in the high word => min = first occ).
// ---------------------------------------------------------------------------
__global__ __launch_bounds__(256) void k_insert(const int* __restrict__ x,
                                                u64* tab, uint32_t mask, int n) {
    __shared__ int tile[1024];
    const int t  = threadIdx.x;
    const int i0 = blockIdx.x * 1024 + t * 4;
    stage4_async(x, n, i0, &tile[t * 4]);

    for (int k = 0; k < 4; ++k) {
        int i = i0 + k;
        if (i >= n) break;
        uint32_t ukey = (uint32_t)tile[t * 4 + k];
        u64 packed = ((u64)(uint32_t)i << 32) | (u64)ukey;
        uint32_t s = hash_key(ukey, mask);
        for (;;) {
            u64 cur = tab[s];
            if (cur != EMPTY64 && (uint32_t)cur == ukey) {
                atomicMin(&tab[s], packed);               // GLOBAL_ATOMIC_MIN_U64
                break;
            }
            if (cur == EMPTY64) {
                u64 prev = atomicCAS(&tab[s], EMPTY64, packed);
                if (prev == EMPTY64) break;               // claimed
                if ((uint32_t)prev == ukey) { atomicMin(&tab[s], packed); break; }
                // different key raced in: fall through and probe on
            }
            s = (s + 1) & mask;
        }
    }
}

// ---------------------------------------------------------------------------
// Flag pass: flag[i] = (first_occurrence(x[i]) == i); per-block flag counts.
// ---------------------------------------------------------------------------
__global__ __launch_bounds__(256) void k_flags(const int* __restrict__ x,
                                               const u64* __restrict__ tab,
                                               uint32_t mask, int n,
                                               uint8_t* __restrict__ flags,
                                               int* __restrict__ blockSums) {
    __shared__ int tile[1024];
    __shared__ int rs[256];
    const int t  = threadIdx.x;
    const int i0 = blockIdx.x * 1024 + t * 4;
    stage4_async(x, n, i0, &tile[t * 4]);

    int cnt = 0;
    for (int k = 0; k < 4; ++k) {
        int i = i0 + k;
        if (i < n) {
            uint32_t f = lookup_val(tab, mask, (uint32_t)tile[t * 4 + k]);
            uint8_t fl = (f == (uint32_t)i) ? (uint8_t)1 : (uint8_t)0;
            flags[i] = fl;
            cnt += fl;
        }
    }
    rs[t] = cnt;
    __syncthreads();
    for (int d = 128; d > 0; d >>= 1) {
        if (t < d) rs[t] += rs[t + d];
        __syncthreads();
    }
    if (t == 0) blockSums[blockIdx.x] = rs[0];
}

// ---------------------------------------------------------------------------
// Exclusive scan of blockSums (single block, chunked Hillis-Steele).
// ---------------------------------------------------------------------------
__global__ __launch_bounds__(1024) void k_scan(const int* __restrict__ sums,
                                               int* __restrict__ offs, int nB) {
    __shared__ int tmp[1024];
    __shared__ int carry;
    const int t = threadIdx.x;
    if (t == 0) carry = 0;
    __syncthreads();
    const int nIter = (nB + 1023) / 1024;
    for (int it = 0; it < nIter; ++it) {
        int idx = it * 1024 + t;
        int v = (idx < nB) ? sums[idx] : 0;
        tmp[t] = v;
        __syncthreads();
        for (int d = 1; d < 1024; d <<= 1) {
            int u = (t >= d) ? tmp[t - d] : 0;
            __syncthreads();
            tmp[t] += u;
            __syncthreads();
        }
        int incl = tmp[t];
        int c = carry;
        __syncthreads();
        if (idx < nB) offs[idx] = c + incl - v;        // exclusive
        if (t == 1023) carry = c + tmp[1023];
        __syncthreads();
    }
}

// ---------------------------------------------------------------------------
// Rank pass: intra-block scan of flags -> global rank of each first
// occurrence; overwrite the key's hash-slot value word with its rank.
// Exactly one writer per slot; the key (low) word is unchanged, so
// concurrent probe chains through the slot remain valid.
// ---------------------------------------------------------------------------
__global__ __launch_bounds__(256) void k_rank(const int* __restrict__ x,
                                              u64* tab, uint32_t mask, int n,
                                              const uint8_t* __restrict__ flags,
                                              const int* __restrict__ blockOffs) {
    __shared__ int tile[1024];
    __shared__ int rs[256];
    const int t  = threadIdx.x;
    const int i0 = blockIdx.x * 1024 + t * 4;
    stage4_async(x, n, i0, &tile[t * 4]);

    int f[4];
    int s = 0;
    for (int k = 0; k < 4; ++k) {
        int i = i0 + k;
        f[k] = (i < n) ? (int)flags[i] : 0;
        s += f[k];
    }
    rs[t] = s;
    __syncthreads();
    for (int d = 1; d < 256; d <<= 1) {
        int u = (t >= d) ? rs[t - d] : 0;
        __syncthreads();
        rs[t] += u;
        __syncthreads();
    }
    int rank = rs[t] - s + blockOffs[blockIdx.x];      // exclusive + block base
    for (int k = 0; k < 4; ++k) {
        int i = i0 + k;
        if (i < n && f[k]) {
            uint32_t ukey = (uint32_t)tile[t * 4 + k];
            uint32_t sl = hash_key(ukey, mask);
            for (uint32_t it = 0; it <= mask; ++it) {
                u64 cur = tab[sl];
                if (cur == EMPTY64) break;             // shouldn't happen
                if ((uint32_t)cur == ukey) {
                    tab[sl] = ((u64)(uint32_t)rank << 32) | (u64)ukey;
                    break;
                }
                sl = (sl + 1) & mask;
            }
            ++rank;
        }
    }
}

// ---------------------------------------------------------------------------
// Emit pass: id = rank+1 if rank < max_tokens else 0; 4 consecutive float
// stores per lane -> coalesced b128 stores.
// ---------------------------------------------------------------------------
__global__ __launch_bounds__(256) void k_emit(const int* __restrict__ x,
                                              const u64* __restrict__ tab,
                                              uint32_t mask, int n,
                                              const int* __restrict__ maxTokPtr,
                                              float* __restrict__ out) {
    __shared__ int tile[1024];
    const uint32_t maxTok = (uint32_t)maxTokPtr[0];
    const int t  = threadIdx.x;
    const int i0 = blockIdx.x * 1024 + t * 4;
    stage4_async(x, n, i0, &tile[t * 4]);

#pragma unroll
    for (int k = 0; k < 4; ++k) {
        int i = i0 + k;
        if (i < n) {
            uint32_t r = lookup_val(tab, mask, (uint32_t)tile[t * 4 + k]); // r = rank
            uint32_t id = (r < maxTok) ? (r + 1u) : 0u;
            out[i] = (float)id;
        }
    }
}

// ---------------------------------------------------------------------------
extern "C" void kernel_launch(void* const* d_in, const int* in_sizes, int n_in,
                              void* d_out, int out_size, void* d_ws, size_t ws_size,
                              hipStream_t stream) {
    const int* x  = (const int*)d_in[0];
    const int* mt = (const int*)d_in[1];
    const int  n  = in_sizes[0];
    float* out = (float*)d_out;

    const int nB = (n + 1023) / 1024;

    // Hash table sizing: 2^22 * 8B = 32MB (L2-resident on MI455X's 192MB L2);
    // deterministically shrink if the workspace is small.
    int hashBits = 22;
    while (hashBits > 18 &&
           ((size_t)(8ull << hashBits) + (size_t)n + (size_t)nB * 8 + 4096) > ws_size)
        --hashBits;
    const uint32_t slots = 1u << hashBits;
    const uint32_t mask  = slots - 1u;

    char* p = (char*)d_ws;
    u64* tab = (u64*)p;             p += (size_t)8 * slots;
    uint8_t* flags = (uint8_t*)p;   p += ((size_t)n + 255) & ~(size_t)255;
    int* sums = (int*)p;            p += ((size_t)nB * 4 + 255) & ~(size_t)255;
    int* offs = (int*)p;

    k_init  <<<dim3(512), dim3(256), 0, stream>>>(tab, slots);
    k_insert<<<dim3(nB),  dim3(256), 0, stream>>>(x, tab, mask, n);
    k_flags <<<dim3(nB),  dim3(256), 0, stream>>>(x, tab, mask, n, flags, sums);
    k_scan  <<<dim3(1),   dim3(1024), 0, stream>>>(sums, offs, nB);
    k_rank  <<<dim3(nB),  dim3(256), 0, stream>>>(x, tab, mask, n, flags, offs);
    k_emit  <<<dim3(nB),  dim3(256), 0, stream>>>(x, tab, mask, n, mt, out);
}